// InversePoseMatrixLayer_72378788873084
// MI455X (gfx1250) — compile-verified
//
#include <hip/hip_runtime.h>

#define THREADS 256
#define POSES_PER_BLOCK 256   // one tile = 256 poses = 12 KB in / 12 KB out

typedef float float4v __attribute__((ext_vector_type(4)));
typedef int   int4v   __attribute__((ext_vector_type(4)));

// ---- CDNA5 async global<->LDS copies (ASYNCcnt path) ----------------------
__device__ __forceinline__ void async_load_b128(const float* gsrc, float* lds_dst) {
#if defined(__has_builtin) && __has_builtin(__builtin_amdgcn_global_load_async_to_lds_b128)
    __builtin_amdgcn_global_load_async_to_lds_b128(
        (__attribute__((address_space(1))) int4v*)gsrc,
        (__attribute__((address_space(3))) int4v*)lds_dst,
        0, 0);
#else
    unsigned int lds_addr = (unsigned int)(unsigned long long)lds_dst;
    unsigned long long gaddr = (unsigned long long)gsrc;
    asm volatile("global_load_async_to_lds_b128 %0, %1, off"
                 :: "v"(lds_addr), "v"(gaddr) : "memory");
#endif
}

__device__ __forceinline__ void async_store_b128(float* gdst, const float* lds_src) {
#if defined(__has_builtin) && __has_builtin(__builtin_amdgcn_global_store_async_from_lds_b128)
    __builtin_amdgcn_global_store_async_from_lds_b128(
        (__attribute__((address_space(1))) int4v*)gdst,
        (__attribute__((address_space(3))) int4v*)lds_src,
        0, 0);
#else
    float4v v = *(const float4v*)lds_src;
    __builtin_nontemporal_store(v, (float4v*)gdst);
#endif
}

__device__ __forceinline__ void wait_async_zero() {
#if defined(__has_builtin) && __has_builtin(__builtin_amdgcn_s_wait_asynccnt)
    __builtin_amdgcn_s_wait_asynccnt(0);
#else
    asm volatile("s_wait_asynccnt 0" ::: "memory");
#endif
}

// ---------------------------------------------------------------------------
// Full-tile kernel: every block owns exactly 256 poses; no guards, no exec
// masking around the async ops -> straight-line wave32 issue stream.
__global__ __launch_bounds__(THREADS)
void inv_pose_full_kernel(const float* __restrict__ x,
                          float* __restrict__ out) {
    __shared__ __align__(16) float s_in [POSES_PER_BLOCK * 12];
    __shared__ __align__(16) float s_out[POSES_PER_BLOCK * 12];

    const int t = threadIdx.x;
    const long long fbase = (long long)blockIdx.x * (POSES_PER_BLOCK * 12);

    // stage 1: contiguous 12 KB tile -> LDS, perfectly coalesced b128 async
    #pragma unroll
    for (int k = 0; k < 3; ++k) {
        const int i4 = t + k * THREADS;
        async_load_b128(x + fbase + i4 * 4, &s_in[i4 * 4]);
    }
    wait_async_zero();   // this wave's async LDS writes done
    __syncthreads();     // whole tile visible to all waves

    // stage 2: per-pose inverse  out = [R^T | -R^T t]
    {
        const float* m = &s_in[t * 12];
        const float m0 = m[0], m1 = m[1], m2  = m[2],  m3  = m[3];
        const float m4 = m[4], m5 = m[5], m6  = m[6],  m7  = m[7];
        const float m8 = m[8], m9 = m[9], m10 = m[10], m11 = m[11];
        float* o = &s_out[t * 12];
        o[0] = m0; o[1] = m4; o[2]  = m8;  o[3]  = -(m0 * m3 + m4 * m7 + m8  * m11);
        o[4] = m1; o[5] = m5; o[6]  = m9;  o[7]  = -(m1 * m3 + m5 * m7 + m9  * m11);
        o[8] = m2; o[9] = m6; o[10] = m10; o[11] = -(m2 * m3 + m6 * m7 + m10 * m11);
    }
    __syncthreads();

    // stage 3: contiguous 12 KB tile LDS -> global, b128 async stores
    // (S_ENDPGM implicitly waits all counters, incl. ASYNCcnt)
    #pragma unroll
    for (int k = 0; k < 3; ++k) {
        const int i4 = t + k * THREADS;
        async_store_b128(out + fbase + i4 * 4, &s_out[i4 * 4]);
    }
}

// Tail kernel: handles n % 256 leftover poses (never taken for B = 4M, but
// keeps kernel_launch correct for any size). Simple direct path.
__global__ __launch_bounds__(THREADS)
void inv_pose_tail_kernel(const float* __restrict__ x,
                          float* __restrict__ out,
                          int first_pose, int n_poses) {
    const int p = first_pose + threadIdx.x;
    if (p >= n_poses) return;
    const float* m = x + (long long)p * 12;
    const float m0 = m[0], m1 = m[1], m2  = m[2],  m3  = m[3];
    const float m4 = m[4], m5 = m[5], m6  = m[6],  m7  = m[7];
    const float m8 = m[8], m9 = m[9], m10 = m[10], m11 = m[11];
    float* o = out + (long long)p * 12;
    o[0] = m0; o[1] = m4; o[2]  = m8;  o[3]  = -(m0 * m3 + m4 * m7 + m8  * m11);
    o[4] = m1; o[5] = m5; o[6]  = m9;  o[7]  = -(m1 * m3 + m5 * m7 + m9  * m11);
    o[8] = m2; o[9] = m6; o[10] = m10; o[11] = -(m2 * m3 + m6 * m7 + m10 * m11);
}

extern "C" void kernel_launch(void* const* d_in, const int* in_sizes, int n_in,
                              void* d_out, int out_size, void* d_ws, size_t ws_size,
                              hipStream_t stream) {
    const float* x = (const float*)d_in[0];
    float* out = (float*)d_out;
    const int n_poses = in_sizes[0] / 12;                 // 4,000,000
    const int n_full  = n_poses / POSES_PER_BLOCK;        // 15,625 full tiles
    const int rem     = n_poses - n_full * POSES_PER_BLOCK; // 0 here

    if (n_full > 0)
        hipLaunchKernelGGL(inv_pose_full_kernel,
                           dim3(n_full), dim3(THREADS), 0, stream, x, out);
    if (rem > 0)
        hipLaunchKernelGGL(inv_pose_tail_kernel,
                           dim3(1), dim3(THREADS), 0, stream,
                           x, out, n_full * POSES_PER_BLOCK, n_poses);
}